// InnerModel_58815282152044
// MI455X (gfx1250) — compile-verified
//
#include <hip/hip_runtime.h>

typedef __attribute__((ext_vector_type(2))) float v2f;
typedef __attribute__((ext_vector_type(4))) float v4f;
typedef __attribute__((ext_vector_type(8))) float v8f;

#define LRELU_ALPHA 0.2f

// ---------------------------------------------------------------------------
// Phase 0: fold W_GDCD through the projections and pack into WMMA B layout.
// Bpack[(k/4)*NC*4 + c*4 + (k&3)] = sum_j P[k][j] * W[wblock*64 + j][ccol]
// (so a wave's B fragment for one 16x16x4 WMMA step is a single b64 load)
// ---------------------------------------------------------------------------
__global__ void fold_kernel(const float* __restrict__ Pd, const float* __restrict__ Pg,
                            const float* __restrict__ Pc, const float* __restrict__ W,
                            float* __restrict__ Bd, float* __restrict__ Bg,
                            float* __restrict__ Bc) {
  int t = blockIdx.x * blockDim.x + threadIdx.x;   // 0 .. 65535
  int k = t >> 8;                                  // 0..255 (K index)
  int c = t & 255;                                 // 0..255 (virtual column)
  const float* P;
  float* dst;
  int wblock, ccol, NC, dstc;
  if (c < 128)      { P = Pd; dst = Bd; wblock = c >> 6; ccol = c & 63;  NC = 128; dstc = c;      }
  else if (c < 192) { P = Pg; dst = Bg; wblock = 2;      ccol = c - 128; NC = 64;  dstc = c - 128; }
  else              { P = Pc; dst = Bc; wblock = 3;      ccol = c - 192; NC = 64;  dstc = c - 192; }
  float acc = 0.0f;
  #pragma unroll 8
  for (int j = 0; j < 64; ++j)
    acc += P[k * 64 + j] * W[(wblock * 64 + j) * 64 + ccol];
  dst[(size_t)(k >> 2) * NC * 4 + dstc * 4 + (k & 3)] = acc;
}

// ---------------------------------------------------------------------------
// Phase 1: node GEMM  G[N,NC] = A[N,256] @ Bpack  via V_WMMA_F32_16X16X4_F32.
// 256 threads = 8 waves; each wave owns a 16-row strip and all NC/16 col tiles
// (A fragment loaded once per K-step, reused across col tiles). A loads use
// the default RT policy: each 128B line of a feats row is consumed across 8
// consecutive K-chunks, so evict-early (NT) would defeat that L0/L1 reuse.
// ---------------------------------------------------------------------------
template <int NC>
__global__ void node_gemm_kernel(const float* __restrict__ A,
                                 const float* __restrict__ Bpack,
                                 float* __restrict__ G, int N) {
  const int lane = threadIdx.x & 31;
  const int wave = threadIdx.x >> 5;
  const int half = lane >> 4;       // 0: K={0,1}, 1: K={2,3} within a step
  const int mcol = lane & 15;       // matrix row (A) / column (B,D)
  const int rowBase = blockIdx.x * 128 + wave * 16;

  int arow = rowBase + mcol;
  if (arow >= N) arow = N - 1;      // clamp: keep EXEC all-1s for WMMA

  const float* aptr = A + (size_t)arow * 256 + half * 2;
  const float* bptr = Bpack + mcol * 4 + half * 2;

  v8f acc[NC / 16] = {};

  for (int k0 = 0; k0 < 256; k0 += 4) {
    v2f afrag = *(const v2f*)(aptr);          // RT: line reused over 8 K-chunks
    aptr += 4;
    #pragma unroll
    for (int j = 0; j < NC / 16; ++j) {
      v2f bfrag = *(const v2f*)(bptr + j * 64);  // hot in cache (128 KB total)
      acc[j] = __builtin_amdgcn_wmma_f32_16x16x4_f32(
          /*neg_a=*/false, afrag, /*neg_b=*/false, bfrag,
          /*c_mod=*/(short)0, acc[j], /*reuse_a=*/false, /*reuse_b=*/false);
    }
    bptr += NC * 4;
  }

  // D layout: VGPR r -> row rowBase + r + half*8, col = mcol
  const int srowBase = rowBase + half * 8;
  #pragma unroll
  for (int j = 0; j < NC / 16; ++j) {
    #pragma unroll
    for (int r = 0; r < 8; ++r) {
      int row = srowBase + r;
      if (row < N) G[(size_t)row * NC + j * 16 + mcol] = acc[j][r];
    }
  }
}

// ---------------------------------------------------------------------------
// Phase 2: edge gather + add + LeakyReLU. Pure bandwidth; 16 lanes per edge,
// float4 per lane (256 B coalesced output per edge). Output is streamed with
// a non-temporal store so the 77 MB G working set stays resident in the
// 192 MB L2 while 1 GB of gather reads replay against it.
// ---------------------------------------------------------------------------
__device__ __forceinline__ float lrelu(float x) {
  return x > 0.0f ? x : LRELU_ALPHA * x;
}

__global__ void edge_kernel(const float* __restrict__ Gd, const float* __restrict__ Gg,
                            const float* __restrict__ Gc,
                            const int* __restrict__ I0, const int* __restrict__ I1,
                            const int* __restrict__ Ig, const int* __restrict__ Ic,
                            float* __restrict__ out, int E) {
  long long t = (long long)blockIdx.x * blockDim.x + threadIdx.x;
  int e = (int)(t >> 4);
  int c = (int)(t & 15) * 4;
  if (e >= E) return;

  v4f a = *(const v4f*)(Gd + (size_t)I0[e] * 128 + c);
  v4f b = *(const v4f*)(Gd + (size_t)I1[e] * 128 + 64 + c);
  v4f g = *(const v4f*)(Gg + (size_t)Ig[e] * 64 + c);
  v4f h = *(const v4f*)(Gc + (size_t)Ic[e] * 64 + c);

  v4f s;
  s.x = lrelu(a.x + b.x + g.x + h.x);
  s.y = lrelu(a.y + b.y + g.y + h.y);
  s.z = lrelu(a.z + b.z + g.z + h.z);
  s.w = lrelu(a.w + b.w + g.w + h.w);

  __builtin_nontemporal_store(s, (v4f*)(out + (size_t)e * 64 + c));
}

// ---------------------------------------------------------------------------
extern "C" void kernel_launch(void* const* d_in, const int* in_sizes, int n_in,
                              void* d_out, int out_size, void* d_ws, size_t ws_size,
                              hipStream_t stream) {
  const float* disease = (const float*)d_in[0];
  const float* gene    = (const float*)d_in[1];
  const float* chem    = (const float*)d_in[2];
  // d_in[3] species_feats: dead code in the reference (GDCD pattern)
  const float* Pd = (const float*)d_in[4];
  const float* Pg = (const float*)d_in[5];
  const float* Pc = (const float*)d_in[6];
  // d_in[7] P_species: unused
  const float* W  = (const float*)d_in[8];
  const int* i0 = (const int*)d_in[9];
  const int* i1 = (const int*)d_in[10];
  const int* ig = (const int*)d_in[11];
  const int* ic = (const int*)d_in[12];
  float* out = (float*)d_out;

  const int N_D = in_sizes[0] / 256;
  const int N_G = in_sizes[1] / 256;
  const int N_C = in_sizes[2] / 256;
  const int E   = in_sizes[9];

  char* ws = (char*)d_ws;
  float* Bd = (float*)(ws);                              // 256*128*4 = 128 KB
  float* Bg = (float*)(ws + 131072);                     // 256*64*4  =  64 KB
  float* Bc = (float*)(ws + 196608);                     // 256*64*4  =  64 KB
  float* Gd = (float*)(ws + 262144);                     // N_D*128*4
  float* Gg = (float*)(ws + 262144 + (size_t)N_D * 512);
  float* Gc = (float*)(ws + 262144 + (size_t)N_D * 512 + (size_t)N_G * 256);

  // Phase 0: fold projections with W blocks (tiny: 65536 dot-64s)
  fold_kernel<<<256, 256, 0, stream>>>(Pd, Pg, Pc, W, Bd, Bg, Bc);

  // Phase 1: node GEMMs via f32 WMMA
  node_gemm_kernel<128><<<(N_D + 127) / 128, 256, 0, stream>>>(disease, Bd, Gd, N_D);
  node_gemm_kernel<64><<<(N_G + 127) / 128, 256, 0, stream>>>(gene, Bg, Gg, N_G);
  node_gemm_kernel<64><<<(N_C + 127) / 128, 256, 0, stream>>>(chem, Bc, Gc, N_C);

  // Phase 2: streaming edge gather + LeakyReLU
  long long threads = (long long)E * 16;
  unsigned blocks = (unsigned)((threads + 255) / 256);
  edge_kernel<<<blocks, 256, 0, stream>>>(Gd, Gg, Gc, i0, i1, ig, ic, out, E);
}